// ClashDetectionGNN_27118423507648
// MI455X (gfx1250) — compile-verified
//
#include <hip/hip_runtime.h>
#include <hip/hip_bf16.h>
#include <math.h>

typedef float v2f __attribute__((ext_vector_type(2)));
typedef float v8f __attribute__((ext_vector_type(8)));

#define NN   20000
#define EEDG 320000
#define ETOT (EEDG + NN)
#define HIDC 128

// ---------------------------------------------------------------------------
// WMMA fp32 GEMM: Out[M,Nc] = A[M,K] @ W[K,Nc]   (M,Nc multiples of 16, K of 4)
// One wave32 per 16x16 output tile, v_wmma_f32_16x16x4_f32 inner loop.
// A fragment: lane l<16 holds A[row][k..k+1], l>=16 holds A[row][k+2..k+3]
// B fragment: lane l<16 holds B[k..k+1][col], l>=16 holds B[k+2..k+3][col]
// C/D: vgpr v -> row v + 8*(lane>=16), col = lane&15
// ---------------------------------------------------------------------------
__global__ __launch_bounds__(128) void gemm_wmma_f32(
    const float* __restrict__ A, const float* __restrict__ W,
    float* __restrict__ Out, int M, int K, int Nc) {
  const int lane   = threadIdx.x & 31;
  const int wave   = threadIdx.x >> 5;
  const int tilesN = Nc >> 4;
  const int tile   = blockIdx.x * 4 + wave;
  if (tile >= (M >> 4) * tilesN) return;           // wave-uniform exit
  const int tM = tile / tilesN;
  const int tN = tile - tM * tilesN;
  const int row   = (tM << 4) + (lane & 15);
  const int col   = (tN << 4) + (lane & 15);
  const int khalf = (lane >> 4) << 1;              // 0 or 2
  const float* __restrict__ Ar = A + (size_t)row * K + khalf;
  const float* __restrict__ Wc = W + (size_t)khalf * Nc + col;
  v8f acc = {};
  for (int k = 0; k < K; k += 4) {
    v2f a, b;
    a.x = Ar[k];
    a.y = Ar[k + 1];
    b.x = Wc[(size_t)k * Nc];
    b.y = Wc[(size_t)(k + 1) * Nc];
    acc = __builtin_amdgcn_wmma_f32_16x16x4_f32(false, a, false, b,
                                                (short)0, acc, false, false);
  }
  const int orow = (tM << 4) + ((lane >> 4) << 3);
  float* __restrict__ Op = Out + (size_t)orow * Nc + col;
#pragma unroll
  for (int v = 0; v < 8; ++v) Op[(size_t)v * Nc] = acc[v];
}

// ---------------------------------------------------------------------------
// Attention coefficients: al_s[n,h] = <h[n,h,:], as[h,:]>, same for al_d.
// One wave per (node, head); wave32 shuffle reduction.
// ---------------------------------------------------------------------------
__global__ __launch_bounds__(128) void att_coef(
    const float* __restrict__ h, const float* __restrict__ as_,
    const float* __restrict__ ad_, float* __restrict__ al_s,
    float* __restrict__ al_d, int Nn, int Hh, int C) {
  const int lane = threadIdx.x & 31;
  const int wave = threadIdx.x >> 5;
  const int idx  = blockIdx.x * 4 + wave;          // flat node*H + head
  if (idx >= Nn * Hh) return;
  const int n = idx / Hh, hd = idx - n * Hh;
  const float* hp = h + (size_t)n * Hh * C + (size_t)hd * C;
  float ss = 0.f, sd = 0.f;
  for (int c = lane; c < C; c += 32) {
    float v = hp[c];
    ss += v * as_[hd * C + c];
    sd += v * ad_[hd * C + c];
  }
#pragma unroll
  for (int o = 16; o > 0; o >>= 1) {
    ss += __shfl_xor(ss, o, 32);
    sd += __shfl_xor(sd, o, 32);
  }
  if (lane == 0) { al_s[idx] = ss; al_d[idx] = sd; }
}

// Monotone float<->uint key mapping for atomicMax-based segment max.
__device__ __forceinline__ unsigned fkey(float f) {
  unsigned u = __float_as_uint(f);
  return u ^ ((unsigned)((int)u >> 31) | 0x80000000u);
}
__device__ __forceinline__ float fromkey(unsigned u) {
  unsigned b = (u & 0x80000000u) ? (u ^ 0x80000000u) : ~u;
  return __uint_as_float(b);
}

// Edge pass 1: segment max of leaky_relu(al_s[src]+al_d[dst]) over dst.
__global__ __launch_bounds__(256) void edge_max(
    const int* __restrict__ ei, const float* __restrict__ al_s,
    const float* __restrict__ al_d, unsigned* __restrict__ mkey,
    int Ee, int Nn, int Hh) {
  const int idx = blockIdx.x * 256 + threadIdx.x;
  if (idx >= (Ee + Nn) * Hh) return;
  const int e = idx / Hh, hd = idx - e * Hh;
  const int s = (e < Ee) ? ei[e]      : (e - Ee);
  const int d = (e < Ee) ? ei[Ee + e] : (e - Ee);
  float v = al_s[s * Hh + hd] + al_d[d * Hh + hd];
  v = (v > 0.f) ? v : 0.2f * v;                    // leaky_relu(0.2)
  atomicMax(&mkey[d * Hh + hd], fkey(v));
}

// Edge pass 2: exp(e - max) per edge, accumulate per-dst denominator.
__global__ __launch_bounds__(256) void edge_exp(
    const int* __restrict__ ei, const float* __restrict__ al_s,
    const float* __restrict__ al_d, const unsigned* __restrict__ mkey,
    float* __restrict__ expb, float* __restrict__ denom,
    int Ee, int Nn, int Hh) {
  const int idx = blockIdx.x * 256 + threadIdx.x;
  if (idx >= (Ee + Nn) * Hh) return;
  const int e = idx / Hh, hd = idx - e * Hh;
  const int s = (e < Ee) ? ei[e]      : (e - Ee);
  const int d = (e < Ee) ? ei[Ee + e] : (e - Ee);
  float v = al_s[s * Hh + hd] + al_d[d * Hh + hd];
  v = (v > 0.f) ? v : 0.2f * v;
  const float ex = expf(v - fromkey(mkey[d * Hh + hd]));
  expb[idx] = ex;
  atomicAdd(&denom[d * Hh + hd], ex);
}

// Edge pass 3: out[dst,h,:] += alpha * h[src,h,:]  (float atomics, L2-resident)
__global__ __launch_bounds__(256) void aggregate(
    const int* __restrict__ ei, const float* __restrict__ h,
    const float* __restrict__ expb, const float* __restrict__ denom,
    float* __restrict__ outb, int Ee, int Nn, int Hh) {
  const size_t idx = (size_t)blockIdx.x * 256 + threadIdx.x;
  const size_t tot = (size_t)(Ee + Nn) * Hh * HIDC;
  if (idx >= tot) return;
  const int c    = (int)(idx % HIDC);
  const size_t r = idx / HIDC;
  const int hd   = (int)(r % Hh);
  const int e    = (int)(r / Hh);
  const int s = (e < Ee) ? ei[e]      : (e - Ee);
  const int d = (e < Ee) ? ei[Ee + e] : (e - Ee);
  const float alpha = expb[(size_t)e * Hh + hd] / (denom[d * Hh + hd] + 1e-16f);
  atomicAdd(&outb[((size_t)d * Hh + hd) * HIDC + c],
            alpha * h[((size_t)s * Hh + hd) * HIDC + c]);
}

// Per-channel BatchNorm statistics (population var, ddof=0).
__global__ __launch_bounds__(256) void bn_stats(
    const float* __restrict__ X, float* __restrict__ mu,
    float* __restrict__ rs, int Nn, int D) {
  const int d = blockIdx.x;
  float s = 0.f, s2 = 0.f;
  for (int n = threadIdx.x; n < Nn; n += 256) {
    float v = X[(size_t)n * D + d];
    s += v; s2 += v * v;
  }
  __shared__ float ssum[256], ssq[256];
  ssum[threadIdx.x] = s; ssq[threadIdx.x] = s2;
  __syncthreads();
  for (int o = 128; o > 0; o >>= 1) {
    if (threadIdx.x < o) {
      ssum[threadIdx.x] += ssum[threadIdx.x + o];
      ssq[threadIdx.x]  += ssq[threadIdx.x + o];
    }
    __syncthreads();
  }
  if (threadIdx.x == 0) {
    float m   = ssum[0] / (float)Nn;
    float var = ssq[0] / (float)Nn - m * m;
    mu[d] = m;
    rs[d] = rsqrtf(var + 1e-5f);
  }
}

// Fused BN affine + ELU, in place. (GAT bias cancels under BN mean-subtract.)
__global__ __launch_bounds__(256) void bn_elu(
    float* __restrict__ X, const float* __restrict__ mu,
    const float* __restrict__ rs, const float* __restrict__ g,
    const float* __restrict__ bb, int Nn, int D) {
  const size_t idx = (size_t)blockIdx.x * 256 + threadIdx.x;
  if (idx >= (size_t)Nn * D) return;
  const int d = (int)(idx % D);
  float v = (X[idx] - mu[d]) * rs[d] * g[d] + bb[d];
  X[idx] = (v > 0.f) ? v : (expf(v) - 1.f);
}

// ---------------------------------------------------------------------------
// Edge MLP: per block, 16 edges. Gather [h[src]|h[dst]] -> LDS (padded),
// then two WMMA fp32 GEMM stages (256->128 relu, 128->64 relu) + final dot.
// ---------------------------------------------------------------------------
#define EE_S 264
#define Z1_S 136
#define Z2_S 72
__global__ __launch_bounds__(256) void edge_mlp(
    const float* __restrict__ h2, const int* __restrict__ ei,
    const float* __restrict__ M1, const float* __restrict__ c1,
    const float* __restrict__ M2, const float* __restrict__ c2,
    const float* __restrict__ M3, const float* __restrict__ c3,
    float* __restrict__ out, int Ee) {
  __shared__ float ee[16 * EE_S];
  __shared__ float z1[16 * Z1_S];
  __shared__ float z2[16 * Z2_S];
  const int tid  = threadIdx.x;
  const int lane = tid & 31;
  const int wave = tid >> 5;
  const int e0   = blockIdx.x * 16;

  { // gather: thread t -> row t/16, 16 contiguous features
    const int r  = tid >> 4;
    const int cb = (tid & 15) << 4;
    const int e  = e0 + r;
    const int node = (cb < 128) ? ei[e] : ei[Ee + e];
    const int base = (cb < 128) ? cb : (cb - 128);
    const float* hp = h2 + (size_t)node * HIDC + base;
#pragma unroll
    for (int j = 0; j < 16; ++j) ee[r * EE_S + cb + j] = hp[j];
  }
  __syncthreads();

  { // z1[16,128] = relu(ee @ M1 + c1); wave w -> cols [16w,16w+16)
    const int col   = (wave << 4) + (lane & 15);
    const int rowl  = lane & 15;
    const int khalf = (lane >> 4) << 1;
    v8f acc = {};
    for (int k = 0; k < 256; k += 4) {
      v2f a, b;
      a.x = ee[rowl * EE_S + k + khalf];
      a.y = ee[rowl * EE_S + k + khalf + 1];
      b.x = M1[(size_t)(k + khalf) * 128 + col];
      b.y = M1[(size_t)(k + khalf + 1) * 128 + col];
      acc = __builtin_amdgcn_wmma_f32_16x16x4_f32(false, a, false, b,
                                                  (short)0, acc, false, false);
    }
    const int r0 = (lane >> 4) << 3;
    const float bias = c1[col];
#pragma unroll
    for (int v = 0; v < 8; ++v) {
      float z = acc[v] + bias;
      z1[(r0 + v) * Z1_S + col] = (z > 0.f) ? z : 0.f;
    }
  }
  __syncthreads();

  if (wave < 4) { // z2[16,64] = relu(z1 @ M2 + c2)
    const int col   = (wave << 4) + (lane & 15);
    const int rowl  = lane & 15;
    const int khalf = (lane >> 4) << 1;
    v8f acc = {};
    for (int k = 0; k < 128; k += 4) {
      v2f a, b;
      a.x = z1[rowl * Z1_S + k + khalf];
      a.y = z1[rowl * Z1_S + k + khalf + 1];
      b.x = M2[(size_t)(k + khalf) * 64 + col];
      b.y = M2[(size_t)(k + khalf + 1) * 64 + col];
      acc = __builtin_amdgcn_wmma_f32_16x16x4_f32(false, a, false, b,
                                                  (short)0, acc, false, false);
    }
    const int r0 = (lane >> 4) << 3;
    const float bias = c2[col];
#pragma unroll
    for (int v = 0; v < 8; ++v) {
      float z = acc[v] + bias;
      z2[(r0 + v) * Z2_S + col] = (z > 0.f) ? z : 0.f;
    }
  }
  __syncthreads();

  if (tid < 16) { // logits = z2 @ M3 + c3
    float s = c3[0];
#pragma unroll 8
    for (int k = 0; k < 64; ++k) s += z2[tid * Z2_S + k] * M3[k];
    out[e0 + tid] = s;
  }
}

// ---------------------------------------------------------------------------
extern "C" void kernel_launch(void* const* d_in, const int* in_sizes, int n_in,
                              void* d_out, int out_size, void* d_ws, size_t ws_size,
                              hipStream_t stream) {
  (void)in_sizes; (void)n_in; (void)out_size; (void)ws_size;
  const float* x    = (const float*)d_in[0];
  const int*   ei   = (const int*)d_in[1];
  const float* W0   = (const float*)d_in[2];
  const float* as0  = (const float*)d_in[3];
  const float* ad0  = (const float*)d_in[4];
  const float* W1   = (const float*)d_in[6];
  const float* as1  = (const float*)d_in[7];
  const float* ad1  = (const float*)d_in[8];
  const float* W2   = (const float*)d_in[10];
  const float* as2  = (const float*)d_in[11];
  const float* ad2  = (const float*)d_in[12];
  const float* bn0g = (const float*)d_in[14];
  const float* bn0b = (const float*)d_in[15];
  const float* bn1g = (const float*)d_in[16];
  const float* bn1b = (const float*)d_in[17];
  const float* bn2g = (const float*)d_in[18];
  const float* bn2b = (const float*)d_in[19];
  const float* M1   = (const float*)d_in[20];
  const float* c1   = (const float*)d_in[21];
  const float* M2   = (const float*)d_in[22];
  const float* c2   = (const float*)d_in[23];
  const float* M3   = (const float*)d_in[24];
  const float* c3   = (const float*)d_in[25];
  float* out = (float*)d_out;

  float* ws   = (float*)d_ws;
  float* bufA = ws;                              // [N,512]
  float* bufB = bufA + (size_t)NN * 512;         // [N,512] (h = X@W)
  float* bufC = bufB + (size_t)NN * 512;         // [N,512]
  float* al_s = bufC + (size_t)NN * 512;         // [N,4]
  float* al_d = al_s + (size_t)NN * 4;           // [N,4]
  unsigned* mkey = (unsigned*)(al_d + (size_t)NN * 4); // [N,4]
  float* denom = (float*)(mkey + (size_t)NN * 4);      // [N,4]
  float* expb  = denom + (size_t)NN * 4;         // [Etot,4]
  float* mu    = expb + (size_t)ETOT * 4;        // [512]
  float* rs    = mu + 512;                       // [512]

  auto layer = [&](const float* Xin, const float* Wl, int K, int Hh,
                   const float* asl, const float* adl,
                   float* hbuf, float* aggbuf,
                   const float* bg, const float* bb) {
    const int D = Hh * HIDC;
    const int tiles = (NN / 16) * (D / 16);
    gemm_wmma_f32<<<(tiles + 3) / 4, 128, 0, stream>>>(Xin, Wl, hbuf, NN, K, D);
    att_coef<<<(NN * Hh + 3) / 4, 128, 0, stream>>>(hbuf, asl, adl, al_s, al_d,
                                                    NN, Hh, HIDC);
    hipMemsetAsync(aggbuf, 0, (size_t)NN * D * sizeof(float), stream);
    hipMemsetAsync(mkey, 0, (size_t)NN * Hh * sizeof(unsigned), stream);
    hipMemsetAsync(denom, 0, (size_t)NN * Hh * sizeof(float), stream);
    const int tot = ETOT * Hh;
    edge_max<<<(tot + 255) / 256, 256, 0, stream>>>(ei, al_s, al_d, mkey,
                                                    EEDG, NN, Hh);
    edge_exp<<<(tot + 255) / 256, 256, 0, stream>>>(ei, al_s, al_d, mkey,
                                                    expb, denom, EEDG, NN, Hh);
    const size_t tot2 = (size_t)ETOT * Hh * HIDC;
    aggregate<<<(unsigned)((tot2 + 255) / 256), 256, 0, stream>>>(
        ei, hbuf, expb, denom, aggbuf, EEDG, NN, Hh);
    bn_stats<<<D, 256, 0, stream>>>(aggbuf, mu, rs, NN, D);
    bn_elu<<<(unsigned)(((size_t)NN * D + 255) / 256), 256, 0, stream>>>(
        aggbuf, mu, rs, bg, bb, NN, D);
  };

  layer(x,    W0, 64,  4, as0, ad0, bufB, bufC, bn0g, bn0b);
  layer(bufC, W1, 512, 4, as1, ad1, bufB, bufA, bn1g, bn1b);
  layer(bufA, W2, 512, 1, as2, ad2, bufB, bufC, bn2g, bn2b);

  edge_mlp<<<EEDG / 16, 256, 0, stream>>>(bufC, ei, M1, c1, M2, c2, M3, c3,
                                          out, EEDG);
}